// VanillaRNN_76364518523247
// MI455X (gfx1250) — compile-verified
//
#include <hip/hip_runtime.h>
#include <math.h>

typedef float v2f __attribute__((ext_vector_type(2)));
typedef float v8f __attribute__((ext_vector_type(8)));

#define RNN_B 256
#define RNN_T 2048
#define RNN_I 256
#define RNN_H 256
// 256 + 4 pad: row stride 260 dwords -> (260 % 64) == 4, so the 16 lanes of a
// fragment load land on LDS banks 0,4,...,60: conflict-free. 260*4 = 1040 B
// row pitch is 16B-aligned, so b128 async chunks stay aligned.
#define PAD_STRIDE 260

// Async global->LDS copy of 16 bytes (CDNA5 GLOBAL_LOAD_ASYNC_TO_LDS_B128,
// ASYNCcnt-tracked, no VGPR round trip). lptr must be an LDS pointer.
__device__ __forceinline__ void async_load_b128(const float* gptr, float* lptr) {
  unsigned lds_addr =
      (unsigned)(size_t)(__attribute__((address_space(3))) float*)lptr;
  unsigned long long gaddr = (unsigned long long)(size_t)gptr;
  asm volatile("global_load_async_to_lds_b128 %0, %1, off"
               :: "v"(lds_addr), "v"(gaddr) : "memory");
}

__device__ __forceinline__ void wait_async_all() {
  asm volatile("s_wait_asynccnt 0x0" ::: "memory");
}

// ---------------------------------------------------------------------------
// Kernel 1: xw[bb,t,j] = sum_i x[bb,t,i] * W[j,i], written into d_out in
// (B,T,H) layout so kernel 2 can read-modify-write it in place.
// x is (B*T, 256) row-major; one WG = 512 threads = 16 waves covers a
// 64-row (bt) x 256-col (j) output tile. Wave w owns j-tile [16w,16w+16)
// for all 4 m-tiles; K loop = 64 x V_WMMA_F32_16X16X4_F32.
// ---------------------------------------------------------------------------
__global__ void __launch_bounds__(512)
rnn_xw_wmma(const float* __restrict__ x, const float* __restrict__ W,
            float* __restrict__ out) {
  extern __shared__ float xs[];                  // 64 x PAD_STRIDE
  const int tid = threadIdx.x;
  const size_t bt0 = (size_t)blockIdx.x * 64;

  // Async-stage the 64x256 x tile into padded LDS: 8 b128 async ops/thread.
  const float* xg = x + bt0 * RNN_I;
  for (int i = tid * 4; i < 64 * RNN_I; i += 512 * 4)
    async_load_b128(xg + i, &xs[(i >> 8) * PAD_STRIDE + (i & 255)]);
  wait_async_all();
  __syncthreads();

  const int lane = tid & 31;
  const int wv   = tid >> 5;                     // 0..15 -> j tile
  const int l15  = lane & 15;
  const int hi   = lane >> 4;                    // half-wave select
  const int j0   = wv * 16;
  const int jcol = j0 + l15;
  const int kA   = hi * 2;                       // A frag: lanes16-31 -> K+2,K+3
  const int kB   = hi;                           // B frag: vgpr0 K/K+1, vgpr1 K+2/K+3
  const float* wrow = W + (size_t)jcol * RNN_I;  // B matrix = W^T, N=j fixed per lane

  v8f acc[4];
  #pragma unroll
  for (int m = 0; m < 4; ++m) acc[m] = {};

  for (int kt = 0; kt < 64; ++kt) {
    const int k0 = kt * 4;
    v2f bf;
    bf.x = wrow[k0 + kB];
    bf.y = wrow[k0 + kB + 2];
    #pragma unroll
    for (int m = 0; m < 4; ++m) {
      v2f af = *(const v2f*)&xs[(m * 16 + l15) * PAD_STRIDE + k0 + kA];
      acc[m] = __builtin_amdgcn_wmma_f32_16x16x4_f32(
          false, af, false, bf, (short)0, acc[m], false, false);
    }
  }

  // C/D layout: vgpr r -> M = r + 8*hi, N = l15.
  #pragma unroll
  for (int m = 0; m < 4; ++m) {
    #pragma unroll
    for (int r = 0; r < 8; ++r) {
      const size_t row = bt0 + (size_t)(m * 16 + r + hi * 8);
      out[row * RNN_H + jcol] = acc[m][r];
    }
  }
}

// ---------------------------------------------------------------------------
// Kernel 2: the recurrence. Column block j0..j0+15 is independent of all
// other columns, so grid = 16 WGs. Each WG keeps all of V (256x260 padded,
// 260 KB) plus hT (16x260) in LDS. Per step: 256x256x16 fp32 matmul as
// 16 waves x 64 WMMA_F32_16X16X4_F32, then tanh(+xw +b), written both to
// d_out (in place over xw) and back into hT for step t+1.
// ---------------------------------------------------------------------------
__global__ void __launch_bounds__(512)
rnn_recurrence_wmma(const float* __restrict__ Vm, const float* __restrict__ bias,
                    float* __restrict__ out) {
  extern __shared__ float smem[];
  float* vs = smem;                        // 256 x PAD_STRIDE: V row-major, padded
  float* hT = smem + 256 * PAD_STRIDE;     // 16 x PAD_STRIDE : hT[n][k] = h[k, j0+n]

  const int tid  = threadIdx.x;
  const int lane = tid & 31;
  const int wv   = tid >> 5;               // 0..15 -> output m tile (batch rows)
  const int l15  = lane & 15;
  const int hi   = lane >> 4;
  const int j0   = blockIdx.x * 16;
  const int n    = l15;
  const int jcol = j0 + n;

  // Async-stage all of V into padded LDS (32 b128 async ops/thread),
  // zero hT (h0 = 0) meanwhile.
  for (int i = tid * 4; i < RNN_H * RNN_H; i += 512 * 4)
    async_load_b128(Vm + i, &vs[(i >> 8) * PAD_STRIDE + (i & 255)]);
  for (int i = tid; i < 16 * PAD_STRIDE; i += 512)
    hT[i] = 0.0f;
  wait_async_all();
  __syncthreads();

  const float bj = bias[jcol];
  const int kA = hi * 2;
  const int kB = hi;
  const int m0 = wv * 16;
  const float* arow = vs + (m0 + l15) * PAD_STRIDE + kA;  // A frag row of V
  const float* brow = hT + n * PAD_STRIDE + kB;           // B frag row of hT

  // Per-lane global addresses of the 8 C-tile elements (fixed bb,j; t varies).
  size_t obase[8];
  #pragma unroll
  for (int r = 0; r < 8; ++r) {
    const int bb = m0 + r + hi * 8;
    obase[r] = (size_t)bb * (RNN_T * RNN_H) + (size_t)jcol;
  }

  for (int t = 0; t < RNN_T; ++t) {
    const size_t toff = (size_t)t * RNN_H;

    // Fetch this step's xw (written by kernel 1) and prefetch next step's.
    float xw[8];
    #pragma unroll
    for (int r = 0; r < 8; ++r) {
      xw[r] = out[obase[r] + toff];
      if (t + 1 < RNN_T)
        __builtin_prefetch(&out[obase[r] + toff + RNN_H], 0, 1);
    }

    // C = V[m-tile,:] @ h[:, j-block]
    v8f c = {};
    for (int kt = 0; kt < 64; ++kt) {
      const int k0 = kt * 4;
      v2f af = *(const v2f*)(arow + k0);
      v2f bf;
      bf.x = brow[k0];
      bf.y = brow[k0 + 2];
      c = __builtin_amdgcn_wmma_f32_16x16x4_f32(
          false, af, false, bf, (short)0, c, false, false);
    }

    __syncthreads();  // all waves done reading hT for step t

    #pragma unroll
    for (int r = 0; r < 8; ++r) {
      const float hv = tanhf(c[r] + xw[r] + bj);
      out[obase[r] + toff] = hv;                     // overwrite xw in place
      hT[n * PAD_STRIDE + m0 + r + hi * 8] = hv;     // h_t for step t+1
    }

    __syncthreads();  // h_t visible to all waves
  }
}

// ---------------------------------------------------------------------------
extern "C" void kernel_launch(void* const* d_in, const int* in_sizes, int n_in,
                              void* d_out, int out_size, void* d_ws, size_t ws_size,
                              hipStream_t stream) {
  (void)in_sizes; (void)n_in; (void)out_size; (void)d_ws; (void)ws_size;
  const float* x  = (const float*)d_in[0];
  const float* W  = (const float*)d_in[1];
  const float* Vm = (const float*)d_in[2];
  const float* b  = (const float*)d_in[3];
  float* out = (float*)d_out;

  const int shm1 = 64 * PAD_STRIDE * (int)sizeof(float);          // 66,560 B
  const int shm2 = (256 + 16) * PAD_STRIDE * (int)sizeof(float);  // 282,880 B < 320 KB
  hipFuncSetAttribute((const void*)rnn_xw_wmma,
                      hipFuncAttributeMaxDynamicSharedMemorySize, shm1);
  hipFuncSetAttribute((const void*)rnn_recurrence_wmma,
                      hipFuncAttributeMaxDynamicSharedMemorySize, shm2);

  dim3 blk(512);
  dim3 g1((RNN_B * RNN_T) / 64);   // 8192 WGs for the input-projection GEMM
  rnn_xw_wmma<<<g1, blk, shm1, stream>>>(x, W, out);

  dim3 g2(RNN_H / 16);             // 16 independent column blocks
  rnn_recurrence_wmma<<<g2, blk, shm2, stream>>>(Vm, b, out);
}